// GCNBlock_33904471835028
// MI455X (gfx1250) — compile-verified
//
#include <hip/hip_runtime.h>
#include <hip/hip_bf16.h>

// ---------------- problem constants (match reference) ----------------
constexpr int N_NODES = 100000;   // 6250 * 16  -> exact 16-row WMMA tiles
constexpr int N_EDGES = 1600000;
constexpr int F_FEAT  = 128;      // 8 * 16     -> exact 16-col WMMA tiles
constexpr int N_GRAPH = 64;
constexpr float GN_EPS = 1e-5f;

typedef __attribute__((ext_vector_type(2))) float v2f;
typedef __attribute__((ext_vector_type(8))) float v8f;

// Guaranteed-hardware fp32 global atomic add (global_atomic_add_f32).
// unsafeAtomicAdd lowers via __builtin_amdgcn_global_atomic_fadd_f32; a plain
// atomicAdd(float*) may conservatively become a CAS loop, which would be a
// multi-x slowdown on the contended h[dst] scatter (avg in-degree 16).
__device__ __forceinline__ void atomAddF(float* p, float v) {
#if defined(__HIP_DEVICE_COMPILE__)
  unsafeAtomicAdd(p, v);
#else
  atomicAdd(p, v);
#endif
}

// ---------------------------------------------------------------------
// 0) zero/seed scratch: deg=0, cnt=0, start=N
// ---------------------------------------------------------------------
__global__ void k_zero(float* __restrict__ deg, int* __restrict__ cnt,
                       int* __restrict__ start) {
  int i = blockIdx.x * blockDim.x + threadIdx.x;
  if (i < N_NODES) deg[i] = 0.0f;
  if (i < N_GRAPH) { cnt[i] = 0; start[i] = N_NODES; }
}

// ---------------------------------------------------------------------
// 1) weighted in-degree over real edges (self-loop +1 folded in later)
// ---------------------------------------------------------------------
__global__ void k_deg(const int* __restrict__ dst, const float* __restrict__ ew,
                      float* __restrict__ deg) {
  int e = blockIdx.x * blockDim.x + threadIdx.x;
  if (e < N_EDGES) atomAddF(&deg[dst[e]], ew[e]);
}

// ---------------------------------------------------------------------
// 2) dis = rsqrt(deg + 1)  (deg+1 >= 1, no zero guard needed)
// ---------------------------------------------------------------------
__global__ void k_dis(const float* __restrict__ deg, float* __restrict__ dis) {
  int n = blockIdx.x * blockDim.x + threadIdx.x;
  if (n < N_NODES) dis[n] = __frsqrt_rn(deg[n] + 1.0f);
}

// ---------------------------------------------------------------------
// 3) xw = X @ W with V_WMMA_F32_16X16X4_F32.
//    One block = 256 threads = 8 waves; block owns 16 rows, wave w owns
//    16-col tile w. K loop: 128 / 4 = 32 wmma ops per wave.
//    Fragment layout (wave32): hi = lane>=16 selects K+2 (A) / M+8 (C).
// ---------------------------------------------------------------------
__global__ void __launch_bounds__(256)
k_gemm_xw(const float* __restrict__ X, const float* __restrict__ Wm,
          float* __restrict__ XW) {
  const int wave = threadIdx.x >> 5;
  const int lane = threadIdx.x & 31;
  const int m    = lane & 15;
  const int hi   = lane >> 4;          // 0: K∈{0,1}/M+0 ; 1: K∈{2,3}/M+8
  const int r0   = blockIdx.x * 16;
  const int c0   = wave * 16;

  v8f acc = {};
  const float* __restrict__ arow = X + (size_t)(r0 + m) * F_FEAT + 2 * hi;
  for (int k = 0; k < F_FEAT; k += 4) {
    // A fragment: A[r0+m][k+2*hi], A[r0+m][k+2*hi+1]  (8B-aligned pair)
    v2f a = *(const v2f*)(arow + k);
    // B fragment: W[k+2*hi][c0+m], W[k+2*hi+1][c0+m]
    v2f b;
    b.x = Wm[(size_t)(k + 2 * hi)     * F_FEAT + c0 + m];
    b.y = Wm[(size_t)(k + 2 * hi + 1) * F_FEAT + c0 + m];
    acc = __builtin_amdgcn_wmma_f32_16x16x4_f32(
        /*neg_a=*/false, a, /*neg_b=*/false, b,
        /*c_mod=*/(short)0, acc, /*reuse_a=*/false, /*reuse_b=*/false);
  }
  // C/D layout: VGPR v, lane -> (M = v + 8*hi, N = m)
  float* __restrict__ out = XW + (size_t)(r0 + 8 * hi) * F_FEAT + c0 + m;
#pragma unroll
  for (int v = 0; v < 8; ++v) out[(size_t)v * F_FEAT] = acc[v];
}

// ---------------------------------------------------------------------
// 4) h[n] = b + dis[n]^2 * xw[n]    (bias + self-loop message)
//    one float4 per thread
// ---------------------------------------------------------------------
__global__ void k_init_h(const float* __restrict__ xw,
                         const float* __restrict__ dis,
                         const float* __restrict__ bias,
                         float* __restrict__ h) {
  int idx = blockIdx.x * blockDim.x + threadIdx.x;     // N * 32 threads
  if (idx >= N_NODES * (F_FEAT / 4)) return;
  int n = idx >> 5;
  int q = idx & 31;
  float d2 = dis[n] * dis[n];
  float4 x = ((const float4*)xw)[(size_t)n * 32 + q];
  float4 b = ((const float4*)bias)[q];
  float4 r;
  r.x = b.x + d2 * x.x;  r.y = b.y + d2 * x.y;
  r.z = b.z + d2 * x.z;  r.w = b.w + d2 * x.w;
  ((float4*)h)[(size_t)n * 32 + q] = r;
}

// ---------------------------------------------------------------------
// 5) edge scatter: one wave per edge, 4 feats per lane, hw fp32 atomics
// ---------------------------------------------------------------------
__global__ void __launch_bounds__(256)
k_edges(const int* __restrict__ src, const int* __restrict__ dst,
        const float* __restrict__ ew, const float* __restrict__ dis,
        const float* __restrict__ xw, float* __restrict__ h) {
  int e    = blockIdx.x * (blockDim.x >> 5) + (threadIdx.x >> 5);
  int lane = threadIdx.x & 31;
  if (e >= N_EDGES) return;
  int s = src[e];
  int d = dst[e];
  const float* __restrict__ xs = xw + (size_t)s * F_FEAT;
  __builtin_prefetch(xs + lane * 4, 0, 0);             // global_prefetch_b8
  float nrm = dis[s] * ew[e] * dis[d];
  float4 msg = ((const float4*)xs)[lane];
  float* __restrict__ hd = h + (size_t)d * F_FEAT + lane * 4;
  atomAddF(hd + 0, nrm * msg.x);
  atomAddF(hd + 1, nrm * msg.y);
  atomAddF(hd + 2, nrm * msg.z);
  atomAddF(hd + 3, nrm * msg.w);
}

// ---------------------------------------------------------------------
// 6) per-graph segment bounds (batch is sorted -> contiguous segments)
// ---------------------------------------------------------------------
__global__ void k_segbounds(const int* __restrict__ batch,
                            int* __restrict__ cnt, int* __restrict__ start) {
  int n = blockIdx.x * blockDim.x + threadIdx.x;
  if (n >= N_NODES) return;
  int g = batch[n];
  atomicAdd(&cnt[g], 1);
  atomicMin(&start[g], n);
}

// ---------------------------------------------------------------------
// 7) GraphNorm + ReLU + segment max. One block (512 thr) per graph:
//    thread = (feature f = tid&127, node-stripe st = tid>>7).
//    3 coalesced streaming passes; LDS reduction across the 4 stripes.
// ---------------------------------------------------------------------
__global__ void __launch_bounds__(512)
k_graphnorm(const float* __restrict__ h, const int* __restrict__ cnt,
            const int* __restrict__ start, const float* __restrict__ gw,
            const float* __restrict__ gb, const float* __restrict__ gms,
            float* __restrict__ h_emb, float* __restrict__ flat) {
  __shared__ float red[512];
  const int g   = blockIdx.x;
  const int tid = threadIdx.x;
  const int f   = tid & 127;
  const int st  = tid >> 7;                // 0..3
  const int s0  = start[g];
  const int c   = cnt[g];
  const float inv = 1.0f / (float)(c > 0 ? c : 1);

  // pass 1: mean[f]
  float acc = 0.0f;
  for (int i = st; i < c; i += 4) acc += h[(size_t)(s0 + i) * F_FEAT + f];
  red[tid] = acc;
  __syncthreads();
  if (st == 0) red[f] = (red[f] + red[f + 128] + red[f + 256] + red[f + 384]) * inv;
  __syncthreads();
  const float sub = red[f] * gms[f];       // mean * mean_scale
  __syncthreads();

  // pass 2: inv-std[f]
  float vacc = 0.0f;
  for (int i = st; i < c; i += 4) {
    float o = h[(size_t)(s0 + i) * F_FEAT + f] - sub;
    vacc += o * o;
  }
  red[tid] = vacc;
  __syncthreads();
  if (st == 0) {
    float var = (red[f] + red[f + 128] + red[f + 256] + red[f + 384]) * inv;
    red[f] = __frsqrt_rn(var + GN_EPS);
  }
  __syncthreads();
  const float istd = red[f];
  __syncthreads();

  // pass 3: normalize + relu + running max
  const float wgt = gw[f];
  const float bia = gb[f];
  float mx = -INFINITY;
  for (int i = st; i < c; i += 4) {
    size_t idx = (size_t)(s0 + i) * F_FEAT + f;
    float o = h[idx] - sub;
    float e = fmaxf(wgt * o * istd + bia, 0.0f);
    h_emb[idx] = e;
    mx = fmaxf(mx, e);
  }
  red[tid] = mx;
  __syncthreads();
  if (st == 0)
    flat[(size_t)g * F_FEAT + f] =
        fmaxf(fmaxf(red[f], red[f + 128]), fmaxf(red[f + 256], red[f + 384]));
}

// ---------------------------------------------------------------------
// 8) tuple tail passthrough: edge_index (as float), edge_weight, batch
// ---------------------------------------------------------------------
__global__ void k_tail(const int* __restrict__ ei, const float* __restrict__ ew,
                       const int* __restrict__ batch, float* __restrict__ out) {
  int i = blockIdx.x * blockDim.x + threadIdx.x;
  const int total = 2 * N_EDGES + N_EDGES + N_NODES;
  if (i >= total) return;
  if (i < 2 * N_EDGES)            out[i] = (float)ei[i];
  else if (i < 3 * N_EDGES)       out[i] = ew[i - 2 * N_EDGES];
  else                            out[i] = (float)batch[i - 3 * N_EDGES];
}

// ---------------------------------------------------------------------
extern "C" void kernel_launch(void* const* d_in, const int* in_sizes, int n_in,
                              void* d_out, int out_size, void* d_ws, size_t ws_size,
                              hipStream_t stream) {
  (void)in_sizes; (void)n_in; (void)out_size; (void)ws_size;
  const float* X     = (const float*)d_in[0];                  // [N,F]
  const int*   ei    = (const int*)d_in[1];                    // [2,E]
  const int*   batch = (const int*)d_in[2];                    // [N]
  const float* ew    = (const float*)d_in[3];                  // [E]
  const float* Wm    = (const float*)d_in[4];                  // [F,F]
  const float* bias  = (const float*)d_in[5];                  // [F]
  const float* gw    = (const float*)d_in[6];                  // [F]
  const float* gb    = (const float*)d_in[7];                  // [F]
  const float* gms   = (const float*)d_in[8];                  // [F]
  const int* src = ei;
  const int* dst = ei + N_EDGES;

  // workspace carve-up (all 16B aligned)
  float* xw    = (float*)d_ws;                                  // N*F
  float* h     = xw + (size_t)N_NODES * F_FEAT;                 // N*F
  float* deg   = h  + (size_t)N_NODES * F_FEAT;                 // N
  float* dis   = deg + N_NODES;                                 // N
  int*   cnt   = (int*)(dis + N_NODES);                         // G
  int*   start = cnt + N_GRAPH;                                 // G

  float* out_hemb = (float*)d_out;                              // N*F
  float* out_flat = out_hemb + (size_t)N_NODES * F_FEAT;        // G*F
  float* out_tail = out_flat + (size_t)N_GRAPH * F_FEAT;        // 3E+N

  k_zero<<<(N_NODES + 255) / 256, 256, 0, stream>>>(deg, cnt, start);
  k_deg<<<(N_EDGES + 255) / 256, 256, 0, stream>>>(dst, ew, deg);
  k_dis<<<(N_NODES + 255) / 256, 256, 0, stream>>>(deg, dis);
  k_gemm_xw<<<N_NODES / 16, 256, 0, stream>>>(X, Wm, xw);
  k_init_h<<<(N_NODES * (F_FEAT / 4) + 255) / 256, 256, 0, stream>>>(xw, dis, bias, h);
  k_edges<<<(N_EDGES + 7) / 8, 256, 0, stream>>>(src, dst, ew, dis, xw, h);
  k_segbounds<<<(N_NODES + 255) / 256, 256, 0, stream>>>(batch, cnt, start);
  k_graphnorm<<<N_GRAPH, 512, 0, stream>>>(h, cnt, start, gw, gb, gms,
                                           out_hemb, out_flat);
  k_tail<<<(3 * N_EDGES + N_NODES + 255) / 256, 256, 0, stream>>>(ei, ew, batch, out_tail);
}